// IoUNet_70806830842594
// MI455X (gfx1250) — compile-verified
//
#include <hip/hip_runtime.h>

// PrRoIPool2D(7,7, spatial_scale=0.5) for fixed shapes:
// features [N=8, C=256, H=56, W=56] f32, rois [R=256, 5] f32 -> out [R, C, 7, 7] f32.
//
// Per (roi, channel):  Out = Iy(7x56) * F(56x56) * Ix^T(56x7), scaled by 1/area.
// Computed as  T = F * Ix^T  (K along W, contiguous -> b64 A-fragment loads),
// then         Out = Iy * T  (T staged through a per-wave LDS pane).
// fp32 WMMA (V_WMMA_F32_16X16X4_F32): exact fp32, and compute is far from the
// bottleneck on MI455X — the 25.7 MB feature map lives in the 192 MB L2 and
// this is an L2-gather-bound workload.

typedef float v2f __attribute__((ext_vector_type(2)));
typedef float v8f __attribute__((ext_vector_type(8)));

#define POOLED 7
#define HH 56
#define WW 56
#define CC 256
#define KSTEPS 14          // 56 / 4 K-steps

__device__ __forceinline__ float tentG(float t) {
    // antiderivative of tent max(0, 1-|x|), with G(-1)=0
    return (t <= 0.0f) ? 0.5f * (t + 1.0f) * (t + 1.0f)
                       : 1.0f - 0.5f * (1.0f - t) * (1.0f - t);
}

__device__ __forceinline__ float tentIntegral(float s, float e, float g) {
    float a = fminf(fmaxf(s - g, -1.0f), 1.0f);
    float b = fminf(fmaxf(e - g, -1.0f), 1.0f);
    return tentG(b) - tentG(a);
}

__global__ __launch_bounds__(256)
void prroi_pool_wmma_kernel(const float* __restrict__ features,
                            const float* __restrict__ rois,
                            float* __restrict__ out) {
    // blockIdx.y = roi, blockIdx.x = 64-channel block. 256 threads = 8 wave32 waves.
    const int r    = blockIdx.y;
    const int tid  = threadIdx.x;
    const int lane = tid & 31;
    const int wave = tid >> 5;
    const int l16  = lane & 15;
    const int half = lane >> 4;     // 0: lanes 0-15, 1: lanes 16-31
    const int o    = half << 1;     // K sub-offset per ISA 16x4 f32 A/B layout

    __shared__ float IyL[16][64];        // padded tent integrals (rows>=7, cols>=56 are 0)
    __shared__ float IxL[16][64];
    __shared__ float TL[8][64][16];      // per-wave intermediate T = F * Ix^T (64 x 16)

    // ---- per-ROI geometry ----
    const float x1 = rois[r * 5 + 1] * 0.5f;
    const float y1 = rois[r * 5 + 2] * 0.5f;
    const float x2 = rois[r * 5 + 3] * 0.5f;
    const float y2 = rois[r * 5 + 4] * 0.5f;
    const float bw = (x2 - x1) * (1.0f / POOLED);
    const float bh = (y2 - y1) * (1.0f / POOLED);
    const float area  = bw * bh;
    const float scale = (area > 0.0f) ? (1.0f / fmaxf(area, 1e-12f)) : 0.0f;
    const int   batch = (int)rois[r * 5 + 0];

    // ---- build padded weight tables in LDS (16*64 entries each; H==W==56) ----
    for (int idx = tid; idx < 16 * 64; idx += 256) {
        const int m = idx >> 6;      // pooled index (q for Ix, p for Iy)
        const int g = idx & 63;      // grid index
        float vx = 0.0f, vy = 0.0f;
        if (m < POOLED && g < WW) {
            const float gf = (float)g;
            const float sx = x1 + (float)m * bw;
            const float sy = y1 + (float)m * bh;
            vx = tentIntegral(sx, sx + bw, gf);
            vy = tentIntegral(sy, sy + bh, gf);
        }
        IxL[m][g] = vx;
        IyL[m][g] = vy;
    }
    __syncthreads();

    // ---- preload stationary fragments into VGPRs (once per ROI) ----
    // Stage 1 B = Ix^T (K=w, N=q): lane holds N=q=l16, v0=Ix[q][4k+o], v1=Ix[q][4k+o+1].
    // Stage 2 A = Iy   (M=p, K=h): lane holds M=p=l16, v0=Iy[p][4k+o], v1=Iy[p][4k+o+1].
    v2f bx[KSTEPS], ay[KSTEPS];
#pragma unroll
    for (int k = 0; k < KSTEPS; ++k) {
        const int kk = 4 * k + o;
        bx[k].x = IxL[l16][kk];
        bx[k].y = IxL[l16][kk + 1];
        ay[k].x = IyL[l16][kk];
        ay[k].y = IyL[l16][kk + 1];
    }

    float* const myT = &TL[wave][0][0];
    const size_t fbase = (size_t)(batch * CC) * (HH * WW);

    // Row indices for the four M-tiles of stage 1 (h = 16*mt + l16).
    // Tile 3 rows 56..63 are clamped in-bounds (branch-free); the garbage lands in
    // T rows h>=56, which stage 2 (K = 0..55 only) never reads.
    const int h0r = l16;
    const int h1r = 16 + l16;
    const int h2r = 32 + l16;
    const int h3r = (48 + l16 < HH) ? (48 + l16) : (HH - 1);

    // ---- 8 channels per wave ----
    const int c0 = blockIdx.x * 64 + wave * 8;
    for (int i = 0; i < 8; ++i) {
        const int c = c0 + i;
        const float* __restrict__ Fc = features + fbase + (size_t)c * (HH * WW);
        __builtin_prefetch(Fc + HH * WW, 0, 1);   // speculative prefetch of next channel

        // ===== stage 1: T(64x16) = F(64x56, row-padded) * Ix^T(56x16) =====
        // A (M=h, K=w): pairs are contiguous in W -> one aligned b64 load per fragment.
        v8f t0 = {}, t1 = {}, t2 = {}, t3 = {};
#pragma unroll
        for (int k = 0; k < KSTEPS; ++k) {
            const int col = 4 * k + o;
            const v2f a0 = *(const v2f*)(Fc + h0r * WW + col);
            const v2f a1 = *(const v2f*)(Fc + h1r * WW + col);
            const v2f a2 = *(const v2f*)(Fc + h2r * WW + col);
            const v2f a3 = *(const v2f*)(Fc + h3r * WW + col);
            t0 = __builtin_amdgcn_wmma_f32_16x16x4_f32(false, a0, false, bx[k],
                                                       (short)0, t0, false, false);
            t1 = __builtin_amdgcn_wmma_f32_16x16x4_f32(false, a1, false, bx[k],
                                                       (short)0, t1, false, false);
            t2 = __builtin_amdgcn_wmma_f32_16x16x4_f32(false, a2, false, bx[k],
                                                       (short)0, t2, false, false);
            t3 = __builtin_amdgcn_wmma_f32_16x16x4_f32(false, a3, false, bx[k],
                                                       (short)0, t3, false, false);
        }

        // ===== spill T to this wave's LDS pane =====
        // C/D layout: vgpr j holds M=16*mt + j + 8*half, N = q = l16.
        {
            const int rb = half * 8;
#pragma unroll
            for (int j = 0; j < 8; ++j) {
                myT[(rb + j) * 16 + l16]      = t0[j];
                myT[(16 + rb + j) * 16 + l16] = t1[j];
                myT[(32 + rb + j) * 16 + l16] = t2[j];
                myT[(48 + rb + j) * 16 + l16] = t3[j];
            }
        }
        // same-wave cross-lane LDS handoff: drain DS ops, fence compiler/scheduler
        asm volatile("s_wait_dscnt 0" ::: "memory");
        __builtin_amdgcn_wave_barrier();

        // ===== stage 2: Out(16x16) = Iy(16x56) * T(56x16) =====
        // B (K=h, N=q): lane holds N=q=l16, v0 = T[4k+o][q], v1 = T[4k+o+1][q].
        v8f oacc = {};
#pragma unroll
        for (int k = 0; k < KSTEPS; ++k) {
            const int h = 4 * k + o;
            v2f bb;
            bb.x = myT[h * 16 + l16];
            bb.y = myT[(h + 1) * 16 + l16];
            oacc = __builtin_amdgcn_wmma_f32_16x16x4_f32(false, ay[k], false, bb,
                                                         (short)0, oacc, false, false);
        }
        asm volatile("" ::: "memory");
        __builtin_amdgcn_wave_barrier();   // keep next iteration's T stores behind these reads

        // ===== write out[r][c][p][q] = Out[p][q] * scale =====
        // D layout: M=p = j + 8*half (only half==0, j<7 valid), N=q = l16 (<7 valid).
        if (half == 0 && l16 < POOLED) {
            float* __restrict__ oc = out + ((size_t)r * CC + c) * (POOLED * POOLED);
#pragma unroll
            for (int j = 0; j < POOLED; ++j) {
                oc[j * POOLED + l16] = oacc[j] * scale;
            }
        }
    }
}

extern "C" void kernel_launch(void* const* d_in, const int* in_sizes, int n_in,
                              void* d_out, int out_size, void* d_ws, size_t ws_size,
                              hipStream_t stream) {
    const float* features = (const float*)d_in[0];
    const float* rois     = (const float*)d_in[1];
    float*       out      = (float*)d_out;

    const int R = in_sizes[1] / 5;          // 256 rois
    dim3 grid(CC / 64, R);                  // (channel-block, roi)
    dim3 block(256);                        // 8 wave32 waves
    prroi_pool_wmma_kernel<<<grid, block, 0, stream>>>(features, rois, out);
}